// PHD_VGG16_24885040513057
// MI455X (gfx1250) — compile-verified
//
#include <hip/hip_runtime.h>
#include <hip/hip_bf16.h>
#include <cstdint>
#include <cstddef>

typedef __attribute__((ext_vector_type(2))) float v2f;
typedef __attribute__((ext_vector_type(8))) float v8f;

#define RC 128   // rows of the reduction dim staged in LDS per chunk (per buffer)
#define MAXP 4   // max 16x16 output tiles per wave

#if __has_builtin(__builtin_amdgcn_global_load_async_to_lds_b32)
#define HAVE_ASYNC_LDS 1
#else
#define HAVE_ASYNC_LDS 0
#endif

__device__ __forceinline__ void wait_async_lds() {
#if HAVE_ASYNC_LDS
#if __has_builtin(__builtin_amdgcn_s_wait_asynccnt)
    __builtin_amdgcn_s_wait_asynccnt(0);
#else
    asm volatile("s_wait_asynccnt 0" ::: "memory");
#endif
#endif
}

// Gather one R-chunk of G[r][n] = x[b, c, table[n*10+k]] (r = c*10+k) into LDS.
// LDS layout pairs K-adjacent rows: element (rl, nl) -> ((rl>>1)*cols + nl)*2 + (rl&1)
// so B fragments are contiguous 64-bit LDS loads.
__device__ __forceinline__ void gather_chunk(
    float* GsBuf, const float* __restrict__ xbase, const int* __restrict__ table,
    int nbase, int cols, int rbase, int rcur, int R, int N, int tid)
{
    for (int e = tid; e < rcur * cols; e += 256) {
        const int rl  = e / cols;
        const int nl  = e - rl * cols;
        const int rg  = rbase + rl;
        const int off = ((rl >> 1) * cols + nl) * 2 + (rl & 1);
        if (rg < R) {
            const int c   = rg / 10;
            const int k   = rg - c * 10;
            const int idx = table[(nbase + nl) * 10 + k];
            const float* gp = xbase + (size_t)c * N + idx;
#if HAVE_ASYNC_LDS
            __builtin_amdgcn_global_load_async_to_lds_b32(
                (__attribute__((address_space(1))) int*)gp,
                (__attribute__((address_space(3))) int*)&GsBuf[off],
                0, 0);
#else
            GsBuf[off] = *gp;
#endif
        } else {
            GsBuf[off] = 0.f;   // zero-pad rows past R (plain ds_store)
        }
    }
}

// D[O x N] = ReLU( W[O x R] * G[R x N] + bias ), per batch b.
// One block: 8 waves, N-tile of NT*16 columns, all O tiles; R chunked through a
// double-buffered LDS tile filled with async global->LDS gathers.
// Matrix core: V_WMMA_F32_16X16X4_F32 (exact fp32 math, f32 accumulate).
__global__ __launch_bounds__(256) void phd_conv_wmma(
    const float* __restrict__ x, const float* __restrict__ w,
    const float* __restrict__ bias, const int* __restrict__ table,
    float* __restrict__ y, int C, int O, int N, int NT, int OW, int PPW)
{
    __shared__ float Gs[2][RC * 32];       // 2 x 16 KB double buffer

    const int R    = C * 10;
    const int Rpad = (R + 3) & ~3;         // K=4 granularity
    const int tid  = threadIdx.x;
    const int wave = tid >> 5;
    const int lane = tid & 31;
    const int ln   = lane & 15;
    const int hi   = lane >> 4;            // half-wave select
    const int klo  = hi * 2;               // A/B K offset per ISA fragment layout
    const int cols = NT * 16;
    const int b    = blockIdx.y;
    const int nbase = blockIdx.x * cols;
    const int nsub  = wave / OW;           // which 16-col sub-tile
    const int ow    = wave % OW;           // base O tile of this wave

    v8f acc[MAXP];
#pragma unroll
    for (int p = 0; p < MAXP; ++p) acc[p] = v8f{};

    const float* xbase = x + (size_t)b * (size_t)C * (size_t)N;   // x is (B,C,N)
    const int nch = (Rpad + RC - 1) / RC;

    // prologue: start gather of chunk 0
    gather_chunk(Gs[0], xbase, table, nbase, cols, 0, min(RC, Rpad), R, N, tid);

    for (int ci = 0; ci < nch; ++ci) {
        wait_async_lds();                  // this wave's async fills done
        __syncthreads();                   // all waves' fills visible

        // kick off next chunk into the other buffer (overlaps with WMMA below)
        if (ci + 1 < nch) {
            const int rb2 = (ci + 1) * RC;
            gather_chunk(Gs[(ci + 1) & 1], xbase, table, nbase, cols,
                         rb2, min(RC, Rpad - rb2), R, N, tid);
        }

        const float* buf = Gs[ci & 1];
        const int rbase = ci * RC;
        const int rcur  = min(RC, Rpad - rbase);

        for (int rs = 0; rs < rcur; rs += 4) {
            const int r0 = rs + klo;       // even
            // B fragment: contiguous pair -> single ds_load_b64
            const v2f bf = *(const v2f*)&buf[((r0 >> 1) * cols + nsub * 16 + ln) * 2];
            const int rg = rbase + r0;
#pragma unroll
            for (int p = 0; p < MAXP; ++p) {
                if (p < PPW) {             // uniform per wave
                    const int o = (ow + OW * p) * 16 + ln;
                    v2f af;                // A 16x4 fragment per ISA layout
                    if (rg + 1 < R) {
                        af = *(const v2f*)&w[(size_t)o * R + rg];   // aligned b64
                    } else {
                        af.x = (rg < R) ? w[(size_t)o * R + rg] : 0.f;
                        af.y = 0.f;
                    }
                    acc[p] = __builtin_amdgcn_wmma_f32_16x16x4_f32(
                        false, af, false, bf, (short)0, acc[p], false, false);
                }
            }
        }
        __syncthreads();                   // all reads done before buffer reuse
    }

    // epilogue: bias + ReLU + store (C/D layout: VGPR j -> M=j / M=j+8)
    const int n = nbase + nsub * 16 + ln;
#pragma unroll
    for (int p = 0; p < MAXP; ++p) {
        if (p < PPW) {
            const int o0 = (ow + OW * p) * 16;
#pragma unroll
            for (int j = 0; j < 8; ++j) {
                const int m = j + hi * 8;
                float v = acc[p][j] + bias[o0 + m];
                v = fmaxf(v, 0.f);
                y[((size_t)b * O + o0 + m) * (size_t)N + n] = v;
            }
        }
    }
}

// out[b,c,i] = max_{j<4} h[b, c, adj[pool[i]*4 + j]]
__global__ void phd_pool(const float* __restrict__ h, const int* __restrict__ adj,
                         const int* __restrict__ pool, float* __restrict__ out,
                         int C, int Nin, int Nout)
{
    const int gid = blockIdx.x * blockDim.x + threadIdx.x;
    const int total = 2 * C * Nout;
    if (gid >= total) return;
    const int i = gid % Nout;
    const int t = gid / Nout;
    const int c = t % C;
    const int b = t / C;
    const int p = pool[i];
    const float* base = h + ((size_t)b * C + c) * (size_t)Nin;
    float m = base[adj[p * 4 + 0]];
    m = fmaxf(m, base[adj[p * 4 + 1]]);
    m = fmaxf(m, base[adj[p * 4 + 2]]);
    m = fmaxf(m, base[adj[p * 4 + 3]]);
    out[gid] = m;
}

extern "C" void kernel_launch(void* const* d_in, const int* in_sizes, int n_in,
                              void* d_out, int out_size, void* d_ws, size_t ws_size,
                              hipStream_t stream) {
    (void)in_sizes; (void)n_in; (void)out_size;

    const float* x = (const float*)d_in[0];
    const int *conv_t[5], *adj_t[5], *pool_t[5];
    for (int l = 0; l < 5; ++l) {
        conv_t[l] = (const int*)d_in[1 + 3 * l];
        adj_t[l]  = (const int*)d_in[2 + 3 * l];
        pool_t[l] = (const int*)d_in[3 + 3 * l];
    }
    const float *W[13], *Bi[13];
    for (int i = 0; i < 13; ++i) {
        W[i]  = (const float*)d_in[16 + 2 * i];
        Bi[i] = (const float*)d_in[17 + 2 * i];
    }

    // ping-pong scratch (max intermediate ~42 MB per buffer)
    size_t half = (ws_size / 2) & ~(size_t)511;
    float* bufA = (float*)d_ws;
    float* bufB = (float*)((char*)d_ws + half);

    static const int LN[5]  = {81920, 20480, 5120, 1280, 320};
    static const int Cs[13] = {3, 64,  64, 128,  128, 256, 256,  256, 512, 512,  512, 512, 512};
    static const int Os[13] = {64, 64, 128, 128, 256, 256, 256,  512, 512, 512,  512, 512, 512};
    static const int NCONV[5] = {2, 2, 3, 3, 3};

    const float* cur = x;
    int pp = 0;      // next ping-pong target
    int li = 0;      // running conv-layer index

    for (int lvl = 0; lvl < 5; ++lvl) {
        const int N = LN[lvl];
        int lastO = 0;
        for (int c = 0; c < NCONV[lvl]; ++c, ++li) {
            const int Ci = Cs[li], Oi = Os[li];
            float* outb = pp ? bufB : bufA;
            pp ^= 1;
            const int OT  = Oi / 16;
            const int OW  = OT < 8 ? OT : 8;
            const int NT  = 8 / OW;
            const int PPW = OT / OW;
            dim3 grid(N / (16 * NT), 2, 1);
            phd_conv_wmma<<<grid, 256, 0, stream>>>(cur, W[li], Bi[li], conv_t[lvl],
                                                    outb, Ci, Oi, N, NT, OW, PPW);
            cur = outb;
            lastO = Oi;
        }
        // pool: N -> N/4
        const int Nout = N / 4;
        float* outb;
        if (lvl == 4) {
            outb = (float*)d_out;
        } else {
            outb = pp ? bufB : bufA;
            pp ^= 1;
        }
        const int total = 2 * lastO * Nout;
        phd_pool<<<(total + 255) / 256, 256, 0, stream>>>(cur, adj_t[lvl], pool_t[lvl],
                                                          outb, lastO, N, Nout);
        cur = outb;
    }
}